// SparsePoolingBase_27762668601798
// MI455X (gfx1250) — compile-verified
//
#include <hip/hip_runtime.h>

#define SP_C 32  // channels, fixed by the reference

// ---------------------------------------------------------------------------
// Hot kernel first (so the disasm snippet shows it).
// One wave32 per output voxel.
//   - 4 edge-groups of 8 lanes; each lane loads a float4 => each group pulls
//     one full 128-B feature row per iteration (global_load_b128).
//   - 2x unrolled: up to 8 feature cachelines + 2 index dwords in flight/wave.
//   - per-group register max accumulation, then shfl_xor(8)+shfl_xor(16)
//     folds the 4 groups; lanes 0..7 emit one coalesced 128-B store.
//   - empty segments stay at -inf -> written as 0 (reference isfinite fixup).
// ---------------------------------------------------------------------------
__global__ __launch_bounds__(256) void sp_pool(
    const float* __restrict__ feat,     // [N_IN, 32]
    const int*   __restrict__ in_map,   // [E]
    const int*   __restrict__ start,    // [n_out+1]
    float*       __restrict__ out,      // [n_out, 32]
    int n_out) {
  const int lane = threadIdx.x & 31;
  const int o    = blockIdx.x * 8 + (threadIdx.x >> 5);   // 8 waves / block
  if (o >= n_out) return;

  const int s     = start[o];
  const int e_end = start[o + 1];

  const int grp = lane >> 3;          // 0..3 : which edge slot of 4
  const int sub = lane & 7;           // 0..7 : float4 slot within the row

  const float NEG_INF = -__builtin_inff();
  float4 acc = make_float4(NEG_INF, NEG_INF, NEG_INF, NEG_INF);

  for (int e = s + grp; e < e_end; e += 8) {
    // keep the (streaming, coalesced) index cacheline ahead
    __builtin_prefetch(in_map + e + 64, 0, 0);

    const int idx0 = in_map[e];       // uniform across the 8-lane group
    const float4 v0 =
        *reinterpret_cast<const float4*>(feat + (size_t)idx0 * SP_C + (sub << 2));

    const int e1 = e + 4;
    if (e1 < e_end) {
      const int idx1 = in_map[e1];
      const float4 v1 =
          *reinterpret_cast<const float4*>(feat + (size_t)idx1 * SP_C + (sub << 2));
      acc.x = fmaxf(acc.x, v1.x);
      acc.y = fmaxf(acc.y, v1.y);
      acc.z = fmaxf(acc.z, v1.z);
      acc.w = fmaxf(acc.w, v1.w);
    }

    acc.x = fmaxf(acc.x, v0.x);
    acc.y = fmaxf(acc.y, v0.y);
    acc.z = fmaxf(acc.z, v0.z);
    acc.w = fmaxf(acc.w, v0.w);
  }

  // fold the 4 edge-groups (wave32 cross-lane max)
  acc.x = fmaxf(acc.x, __shfl_xor(acc.x, 8, 32));
  acc.y = fmaxf(acc.y, __shfl_xor(acc.y, 8, 32));
  acc.z = fmaxf(acc.z, __shfl_xor(acc.z, 8, 32));
  acc.w = fmaxf(acc.w, __shfl_xor(acc.w, 8, 32));
  acc.x = fmaxf(acc.x, __shfl_xor(acc.x, 16, 32));
  acc.y = fmaxf(acc.y, __shfl_xor(acc.y, 16, 32));
  acc.z = fmaxf(acc.z, __shfl_xor(acc.z, 16, 32));
  acc.w = fmaxf(acc.w, __shfl_xor(acc.w, 16, 32));

  if (grp == 0) {
    float4 r;
    r.x = (acc.x > NEG_INF && acc.x < -NEG_INF) ? acc.x : 0.0f;
    r.y = (acc.y > NEG_INF && acc.y < -NEG_INF) ? acc.y : 0.0f;
    r.z = (acc.z > NEG_INF && acc.z < -NEG_INF) ? acc.z : 0.0f;
    r.w = (acc.w > NEG_INF && acc.w < -NEG_INF) ? acc.w : 0.0f;
    *reinterpret_cast<float4*>(out + (size_t)o * SP_C + (sub << 2)) = r;
  }
}

// ---------------------------------------------------------------------------
// Kernel 1: CSR row offsets from the sorted out_map via parallel binary search.
// start[o] = first edge e with out_map[e] >= o;  segment(o) = [start[o], start[o+1])
// ---------------------------------------------------------------------------
__global__ __launch_bounds__(256) void sp_lower_bounds(
    const int* __restrict__ out_map, int* __restrict__ start, int E, int n_out) {
  int o = blockIdx.x * blockDim.x + threadIdx.x;
  if (o > n_out) return;
  int lo = 0, hi = E;
  while (lo < hi) {
    int mid = (lo + hi) >> 1;
    if (out_map[mid] < o) lo = mid + 1;
    else                  hi = mid;
  }
  start[o] = lo;
}

// ---------------------------------------------------------------------------
// Host launcher
//   d_in[0]: features f32 [N_IN*32]     (in_sizes[0] = N_IN*32)
//   d_in[1]: in_map  i32 [E]            (in_sizes[1] = E)
//   d_in[2]: out_map i32 [E]  (sorted)
//   d_in[3]: n_out   i32 [1]  (device scalar; host value derived from out_size)
//   d_out  : f32 [n_out*32]
//   d_ws   : >= (n_out+1)*4 bytes for CSR offsets
// ---------------------------------------------------------------------------
extern "C" void kernel_launch(void* const* d_in, const int* in_sizes, int n_in,
                              void* d_out, int out_size, void* d_ws, size_t ws_size,
                              hipStream_t stream) {
  const float* feat    = (const float*)d_in[0];
  const int*   in_map  = (const int*)d_in[1];
  const int*   out_map = (const int*)d_in[2];
  (void)n_in; (void)ws_size;

  const int E     = in_sizes[1];
  const int n_out = out_size / SP_C;

  int* start = (int*)d_ws;  // (n_out + 1) ints

  // CSR row offsets
  {
    const int threads = 256;
    const int total   = n_out + 1;
    const int blocks  = (total + threads - 1) / threads;
    sp_lower_bounds<<<blocks, threads, 0, stream>>>(out_map, start, E, n_out);
  }

  // pooled max, one wave per output voxel (8 waves per block)
  {
    const int threads = 256;
    const int blocks  = (n_out + 7) / 8;
    sp_pool<<<blocks, threads, 0, stream>>>(feat, in_map, start, (float*)d_out, n_out);
  }
}